// DecoderStepWrapper_51745765982790
// MI455X (gfx1250) — compile-verified
//
#include <hip/hip_runtime.h>
#include <hip/hip_bf16.h>
#include <math.h>

typedef __attribute__((ext_vector_type(2))) float v2f;
typedef __attribute__((ext_vector_type(8))) float v8f;

#define LAYERS 28
#define DMODEL 1024
#define NHQ    16
#define NHKV   8
#define HDIM   128
#define GRP    (NHQ / NHKV)      // 2
#define FFDIM  3072
#define VOCAB  151936
#define THETA  1000000.0f
#define EPS    1e-6f
#define BATCH  2
#define PASTN  1024
#define SEQN   (PASTN + 1)       // 1025
#define QSCALE 0.08838834764831845f   // 128^-0.5

// ---------------------------------------------------------------------------
// Bulk copy past KV [L,B,HKV,PAST,HD] -> present [L,B,HKV,PAST+1,HD] (rows
// 0..PAST-1). float4 grid-stride.
// ---------------------------------------------------------------------------
__global__ void copy_past_to_present(const float4* __restrict__ src,
                                     float4* __restrict__ dst, long long n4) {
  const long long srow = (long long)(PASTN * HDIM) / 4;
  const long long drow = (long long)(SEQN * HDIM) / 4;
  long long i = (long long)blockIdx.x * blockDim.x + threadIdx.x;
  long long stride = (long long)gridDim.x * blockDim.x;
  for (; i < n4; i += stride) {
    long long outer = i / srow;
    long long rem   = i - outer * srow;
    dst[outer * drow + rem] = src[i];
  }
}

// cos/sin table: cs[b][0][i]=cos, cs[b][1][i]=sin, i<HD/2. 128 threads.
__global__ void rope_setup(const int* __restrict__ pos_ids, float* __restrict__ cs) {
  int t = threadIdx.x;            // 0..127
  int b = t >> 6, i = t & 63;
  float pos = (float)pos_ids[b];
  float inv = powf(THETA, -((float)(2 * i)) / (float)HDIM);
  float ang = pos * inv;
  cs[(b * 2 + 0) * 64 + i] = cosf(ang);
  cs[(b * 2 + 1) * 64 + i] = sinf(ang);
}

__global__ void init_hidden(const float* __restrict__ e, float* __restrict__ h) {
  int i = blockIdx.x * blockDim.x + threadIdx.x;
  if (i < BATCH * DMODEL) h[i] = e[i];
}

__global__ void zero_buf(float* __restrict__ p, int n) {
  int i = blockIdx.x * blockDim.x + threadIdx.x;
  if (i < n) p[i] = 0.f;
}

// RMSNorm over last dim; one block per batch row.
__global__ __launch_bounds__(256)
void rmsnorm_kernel(const float* __restrict__ x, const float* __restrict__ w,
                    float* __restrict__ out, int n) {
  __shared__ float red[256];
  int b = blockIdx.x;
  const float* xb = x + (long long)b * n;
  float ss = 0.f;
  for (int i = threadIdx.x; i < n; i += 256) { float v = xb[i]; ss += v * v; }
  red[threadIdx.x] = ss;
  __syncthreads();
  for (int s = 128; s > 0; s >>= 1) {
    if (threadIdx.x < s) red[threadIdx.x] += red[threadIdx.x + s];
    __syncthreads();
  }
  float r = rsqrtf(red[0] / (float)n + EPS);
  for (int i = threadIdx.x; i < n; i += 256)
    out[(long long)b * n + i] = xb[i] * r * w[i];
}

// ---------------------------------------------------------------------------
// WMMA GEMV: y[2,N] = x[2,K] @ W[K,N], f32, via V_WMMA_F32_16X16X4_F32.
//  - M padded to 16: rows 2..15 alias row 0 and their C rows are never stored,
//    so control flow is uniform and EXEC is all-ones at every WMMA.
//  - KDIM/NDIM are compile-time so unrolled weight loads use immediate
//    instruction offsets (no per-iteration 64-bit address math).
//  - x chunk staged in LDS: A-fragment reads go through the DS pipe, leaving
//    the VMEM pipe fully for the non-temporal weight stream.
//  - SPLIT-K: blockIdx.y selects a K-chunk; partial sums combine with
//    global_atomic_add_f32 (MODE==1; y pre-initialized with residual or 0).
// ---------------------------------------------------------------------------
template <int KDIM, int NDIM, int SPLIT, int MODE>
__global__ __launch_bounds__(128)
void gemv16_wmma(const float* __restrict__ x, const float* __restrict__ W,
                 float* __restrict__ y) {
  constexpr int KCHUNK = KDIM / SPLIT;
  __shared__ __align__(16) float xs[BATCH][KCHUNK];

  const int kbase = blockIdx.y * KCHUNK;
  for (int i = threadIdx.x; i < BATCH * KCHUNK; i += 128) {
    int bb = i / KCHUNK, kk = i - bb * KCHUNK;
    xs[bb][kk] = x[bb * KDIM + kbase + kk];
  }
  __syncthreads();

  const int lane = threadIdx.x & 31;
  const int wave = threadIdx.x >> 5;
  const int n0   = (blockIdx.x * 4 + wave) * 16;   // NDIM % 64 == 0 always
  const int m    = lane & 15;
  const int r    = lane >> 4;                      // 0 or 1 (half-wave)
  const int koff = r << 1;                         // A: VGPR0 = K+0 / K+2 halves

  const float* __restrict__ xr  = xs[m < BATCH ? m : 0];  // rows 2..15 alias row 0
  const float* __restrict__ pb0 = W + (long long)(kbase + r) * NDIM + n0 + m;
  const float* __restrict__ pb1 = W + (long long)(kbase + 2 + r) * NDIM + n0 + m;

  v8f acc = {};
#pragma unroll 4
  for (int k = 0; k < KCHUNK; k += 4) {
    v2f a, bm;
    a.x  = xr[k + koff];
    a.y  = xr[k + koff + 1];
    bm.x = __builtin_nontemporal_load(pb0 + (long long)k * NDIM);  // rows k, k+1
    bm.y = __builtin_nontemporal_load(pb1 + (long long)k * NDIM);  // rows k+2, k+3
    __builtin_prefetch(pb0 + (long long)(k + 16) * NDIM, 0, 1);
    acc = __builtin_amdgcn_wmma_f32_16x16x4_f32(
        /*neg_a=*/false, a, /*neg_b=*/false, bm,
        /*c_mod=*/(short)0, acc, /*reuse_a=*/false, /*reuse_b=*/false);
  }

  // C layout: VGPR j, lanes 0..15 hold M=j; we need M=0 (b0) and M=1 (b1).
  const int n = n0 + m;
  if (lane < 16) {
    if (MODE == 0) {
      y[n]                   = acc[0];
      y[(long long)NDIM + n] = acc[1];
    } else {
      atomicAdd(&y[n], acc[0]);
      atomicAdd(&y[(long long)NDIM + n], acc[1]);
    }
  }
}

// ---------------------------------------------------------------------------
// Per-head QK RMSNorm + RoPE. Grid = B*(HQ + 2*HKV); block = 128 (= HD).
// ---------------------------------------------------------------------------
__global__ __launch_bounds__(128)
void qk_rope_append(float* __restrict__ qbuf, const float* __restrict__ kbuf,
                    const float* __restrict__ vbuf,
                    const float* __restrict__ qw, const float* __restrict__ kw,
                    const float* __restrict__ cs,
                    float* __restrict__ presK, float* __restrict__ presV) {
  __shared__ float buf[HDIM];
  __shared__ float red[HDIM];
  const int per = NHQ + 2 * NHKV;      // 32 blocks per batch
  int blk = blockIdx.x;
  int b = blk / per;
  int rr = blk - b * per;
  int d = threadIdx.x;
  const float* cosb = cs + (b * 2 + 0) * 64;
  const float* sinb = cs + (b * 2 + 1) * 64;
  float c = cosb[d & 63], sn = sinb[d & 63];

  if (rr < NHQ) {
    int h = rr;
    float v = qbuf[((long long)b * NHQ + h) * HDIM + d];
    red[d] = v * v; __syncthreads();
    for (int s = 64; s > 0; s >>= 1) { if (d < s) red[d] += red[d + s]; __syncthreads(); }
    float rs = rsqrtf(red[0] / (float)HDIM + EPS);
    buf[d] = v * rs * qw[d]; __syncthreads();
    float rot = (d < 64) ? -buf[d + 64] : buf[d - 64];
    qbuf[((long long)b * NHQ + h) * HDIM + d] = buf[d] * c + rot * sn;
  } else if (rr < NHQ + NHKV) {
    int h = rr - NHQ;
    float v = kbuf[((long long)b * NHKV + h) * HDIM + d];
    red[d] = v * v; __syncthreads();
    for (int s = 64; s > 0; s >>= 1) { if (d < s) red[d] += red[d + s]; __syncthreads(); }
    float rs = rsqrtf(red[0] / (float)HDIM + EPS);
    buf[d] = v * rs * kw[d]; __syncthreads();
    float rot = (d < 64) ? -buf[d + 64] : buf[d - 64];
    presK[((long long)(b * NHKV + h)) * SEQN * HDIM + (long long)PASTN * HDIM + d] =
        buf[d] * c + rot * sn;
  } else {
    int h = rr - NHQ - NHKV;
    presV[((long long)(b * NHKV + h)) * SEQN * HDIM + (long long)PASTN * HDIM + d] =
        vbuf[((long long)b * NHKV + h) * HDIM + d];
  }
}

// ---------------------------------------------------------------------------
// GQA attention: one block per (b, kv-head). Scores over SEQ=1025 positions
// for G=2 grouped queries, LDS softmax, then w @ V with coalesced V reads.
// ---------------------------------------------------------------------------
__global__ __launch_bounds__(256)
void attn_kernel(const float* __restrict__ qbuf, const float* __restrict__ presK,
                 const float* __restrict__ presV, float* __restrict__ obuf) {
  __shared__ float sc[GRP][SEQN];
  __shared__ __align__(16) float qs[GRP][HDIM];
  __shared__ float red[256];
  int blk = blockIdx.x;
  int b = blk / NHKV, h = blk - b * NHKV;
  int t = threadIdx.x;

  for (int i = t; i < GRP * HDIM; i += 256) {
    int g = i / HDIM, d = i % HDIM;
    qs[g][d] = qbuf[((long long)b * NHQ + h * GRP + g) * HDIM + d];
  }
  __syncthreads();

  const float* Kh = presK + ((long long)(b * NHKV + h)) * SEQN * HDIM;
  const float* Vh = presV + ((long long)(b * NHKV + h)) * SEQN * HDIM;

  // scores
  for (int s = t; s < SEQN; s += 256) {
    const float4* krow = (const float4*)(Kh + (long long)s * HDIM);
    const float4* q0p = (const float4*)qs[0];
    const float4* q1p = (const float4*)qs[1];
    float d0 = 0.f, d1 = 0.f;
#pragma unroll 8
    for (int i = 0; i < HDIM / 4; ++i) {
      float4 kv = krow[i];
      float4 q0 = q0p[i], q1 = q1p[i];
      d0 += kv.x * q0.x + kv.y * q0.y + kv.z * q0.z + kv.w * q0.w;
      d1 += kv.x * q1.x + kv.y * q1.y + kv.z * q1.z + kv.w * q1.w;
    }
    sc[0][s] = d0 * QSCALE;
    sc[1][s] = d1 * QSCALE;
  }
  __syncthreads();

  // softmax per group
  for (int g = 0; g < GRP; ++g) {
    float mx = -1e30f;
    for (int s = t; s < SEQN; s += 256) mx = fmaxf(mx, sc[g][s]);
    red[t] = mx; __syncthreads();
    for (int s2 = 128; s2 > 0; s2 >>= 1) {
      if (t < s2) red[t] = fmaxf(red[t], red[t + s2]);
      __syncthreads();
    }
    mx = red[0]; __syncthreads();
    float sum = 0.f;
    for (int s = t; s < SEQN; s += 256) {
      float e = __expf(sc[g][s] - mx);
      sc[g][s] = e;
      sum += e;
    }
    red[t] = sum; __syncthreads();
    for (int s2 = 128; s2 > 0; s2 >>= 1) {
      if (t < s2) red[t] += red[t + s2];
      __syncthreads();
    }
    float inv = 1.f / red[0]; __syncthreads();
    for (int s = t; s < SEQN; s += 256) sc[g][s] *= inv;
    __syncthreads();
  }

  // output: thread (g,d) accumulates over all positions; d is coalesced.
  int g = t >> 7, d = t & 127;
  float o = 0.f;
  for (int s = 0; s < SEQN; ++s) o += sc[g][s] * Vh[(long long)s * HDIM + d];
  obuf[((long long)b * NHQ + h * GRP + g) * HDIM + d] = o;
}

__global__ void swiglu_kernel(float* __restrict__ gbuf, const float* __restrict__ ubuf, int n) {
  int i = blockIdx.x * blockDim.x + threadIdx.x;
  if (i < n) {
    float g = gbuf[i];
    float s = g / (1.f + __expf(-g));
    gbuf[i] = s * ubuf[i];
  }
}

// ---------------------------------------------------------------------------
extern "C" void kernel_launch(void* const* d_in, const int* in_sizes, int n_in,
                              void* d_out, int out_size, void* d_ws, size_t ws_size,
                              hipStream_t stream) {
  const float* input_embeds = (const float*)d_in[0];
  const int*   position_ids = (const int*)d_in[1];
  const float* past_keys    = (const float*)d_in[2];
  const float* past_values  = (const float*)d_in[3];
  const float* Wq  = (const float*)d_in[4];
  const float* Wk  = (const float*)d_in[5];
  const float* Wv  = (const float*)d_in[6];
  const float* Wo  = (const float*)d_in[7];
  const float* Wg  = (const float*)d_in[8];
  const float* Wu  = (const float*)d_in[9];
  const float* Wd  = (const float*)d_in[10];
  const float* ln1 = (const float*)d_in[11];
  const float* ln2 = (const float*)d_in[12];
  const float* qn  = (const float*)d_in[13];
  const float* kn  = (const float*)d_in[14];
  const float* fin = (const float*)d_in[15];
  const float* lmh = (const float*)d_in[16];

  float* out    = (float*)d_out;
  float* logits = out;
  float* presK  = out + (size_t)BATCH * VOCAB;
  float* presV  = presK + (size_t)LAYERS * BATCH * NHKV * SEQN * HDIM;

  float* ws = (float*)d_ws;
  float* h   = ws;                          // [B, D]
  float* nrm = h   + BATCH * DMODEL;        // [B, D]
  float* qb  = nrm + BATCH * DMODEL;        // [B, HQ*HD]
  float* kb  = qb  + BATCH * NHQ * HDIM;    // [B, HKV*HD]   (contiguous with qb)
  float* vb  = kb  + BATCH * NHKV * HDIM;   // [B, HKV*HD]   (contiguous)
  float* ob  = vb  + BATCH * NHKV * HDIM;   // [B, HQ*HD]
  float* gb  = ob  + BATCH * NHQ * HDIM;    // [B, FF]
  float* ub  = gb  + BATCH * FFDIM;         // [B, FF]       (contiguous with gb)
  float* cs  = ub  + BATCH * FFDIM;         // [B, 2, 64]

  const int qkvN = BATCH * (NHQ + 2 * NHKV) * HDIM;  // qb+kb+vb floats
  const int guN  = 2 * BATCH * FFDIM;                // gb+ub floats

  // Bulk KV copy into present (independent of compute; stream-ordered first).
  long long n4 = (long long)LAYERS * BATCH * NHKV * PASTN * HDIM / 4;
  copy_past_to_present<<<2048, 256, 0, stream>>>((const float4*)past_keys,   (float4*)presK, n4);
  copy_past_to_present<<<2048, 256, 0, stream>>>((const float4*)past_values, (float4*)presV, n4);
  rope_setup<<<1, 128, 0, stream>>>(position_ids, cs);
  init_hidden<<<(BATCH * DMODEL + 255) / 256, 256, 0, stream>>>(input_embeds, h);

  const size_t kvLayer = (size_t)BATCH * NHKV * SEQN * HDIM;
  for (int l = 0; l < LAYERS; ++l) {
    const float* wq = Wq + (size_t)l * DMODEL * NHQ * HDIM;
    const float* wk = Wk + (size_t)l * DMODEL * NHKV * HDIM;
    const float* wv = Wv + (size_t)l * DMODEL * NHKV * HDIM;
    const float* wo = Wo + (size_t)l * NHQ * HDIM * DMODEL;
    const float* wg = Wg + (size_t)l * DMODEL * FFDIM;
    const float* wu = Wu + (size_t)l * DMODEL * FFDIM;
    const float* wd = Wd + (size_t)l * FFDIM * DMODEL;
    float* pkl = presK + (size_t)l * kvLayer;
    float* pvl = presV + (size_t)l * kvLayer;

    // --- attention block ---
    rmsnorm_kernel<<<BATCH, 256, 0, stream>>>(h, ln1 + (size_t)l * DMODEL, nrm, DMODEL);
    zero_buf<<<(qkvN + 255) / 256, 256, 0, stream>>>(qb, qkvN);
    gemv16_wmma<DMODEL, NHQ * HDIM, 4, 1>
        <<<dim3((NHQ * HDIM) / 64, 4), 128, 0, stream>>>(nrm, wq, qb);
    gemv16_wmma<DMODEL, NHKV * HDIM, 4, 1>
        <<<dim3((NHKV * HDIM) / 64, 4), 128, 0, stream>>>(nrm, wk, kb);
    gemv16_wmma<DMODEL, NHKV * HDIM, 4, 1>
        <<<dim3((NHKV * HDIM) / 64, 4), 128, 0, stream>>>(nrm, wv, vb);
    qk_rope_append<<<BATCH * (NHQ + 2 * NHKV), 128, 0, stream>>>(
        qb, kb, vb, qn + (size_t)l * HDIM, kn + (size_t)l * HDIM, cs, pkl, pvl);
    attn_kernel<<<BATCH * NHKV, 256, 0, stream>>>(qb, pkl, pvl, ob);
    // residual fused: h += ob @ Wo via atomic accumulate onto h
    gemv16_wmma<NHQ * HDIM, DMODEL, 4, 1>
        <<<dim3(DMODEL / 64, 4), 128, 0, stream>>>(ob, wo, h);

    // --- MLP block ---
    rmsnorm_kernel<<<BATCH, 256, 0, stream>>>(h, ln2 + (size_t)l * DMODEL, nrm, DMODEL);
    zero_buf<<<(guN + 255) / 256, 256, 0, stream>>>(gb, guN);
    gemv16_wmma<DMODEL, FFDIM, 2, 1>
        <<<dim3(FFDIM / 64, 2), 128, 0, stream>>>(nrm, wg, gb);
    gemv16_wmma<DMODEL, FFDIM, 2, 1>
        <<<dim3(FFDIM / 64, 2), 128, 0, stream>>>(nrm, wu, ub);
    swiglu_kernel<<<(BATCH * FFDIM + 255) / 256, 256, 0, stream>>>(gb, ub, BATCH * FFDIM);
    gemv16_wmma<FFDIM, DMODEL, 4, 1>
        <<<dim3(DMODEL / 64, 4), 128, 0, stream>>>(gb, wd, h);
  }

  // final norm + LM head (plain-store GEMV, 2374 tile-blocks)
  rmsnorm_kernel<<<BATCH, 256, 0, stream>>>(h, fin, nrm, DMODEL);
  gemv16_wmma<DMODEL, VOCAB, 1, 0>
      <<<dim3(VOCAB / 64, 1), 128, 0, stream>>>(nrm, lmh, logits);
}